// PlanarQuantMSE_38190849196140
// MI455X (gfx1250) — compile-verified
//
#include <hip/hip_runtime.h>
#include <math.h>

// ---------------------------------------------------------------------------
// Problem constants (from reference): N=2048 rows, D=1024, K=256 centroids,
// NG=512 planar rotation groups, d_padded == D (no padding).
// ---------------------------------------------------------------------------
#define PQ_N   2048
#define PQ_D   1024
#define PQ_K   256
#define PQ_NG  512

typedef __attribute__((ext_vector_type(2))) float v2f;
typedef __attribute__((ext_vector_type(8))) float v8f;

// ---------------------------------------------------------------------------
// Kernel 1: row sum-of-squares via FP32 WMMA (V_WMMA_F32_16X16X4_F32).
//   One wave32 per block, 16 rows per wave.
//   A-fragment (16x4 f32, ISA layout): lane L holds row (L&15),
//     K-pair base (L>>4)*2 -> a contiguous float2 from that row.
//   We square A element-wise and use B == ones, so
//     D[i, n] = sum_k x[i,k]^2  (replicated across all 16 columns).
//   C/D layout: VGPR r, lanes 0-15 -> row r; lanes 16-31 -> row 8+r.
// ---------------------------------------------------------------------------
__global__ __launch_bounds__(32) void PQ_norms_wmma(
    const float* __restrict__ x, float* __restrict__ norms) {
  const int lane    = threadIdx.x;          // 0..31
  const int rowBase = blockIdx.x * 16;      // 128 blocks * 16 rows = 2048
  const int row     = rowBase + (lane & 15);
  const int kb      = (lane >> 4) * 2;      // which K pair this lane feeds

  const float* p = x + (size_t)row * PQ_D + kb;

  v8f acc = {};
  v2f ones;
  ones[0] = 1.0f; ones[1] = 1.0f;

#pragma unroll 8
  for (int k = 0; k < PQ_D; k += 4) {
    v2f a = *(const v2f*)(p + k);
    a = a * a;  // element-wise square -> Gram trick with B = ones
    // 8 args: (neg_a, A, neg_b, B, c_mod, C, reuse_a, reuse_b)
    acc = __builtin_amdgcn_wmma_f32_16x16x4_f32(
        false, a, false, ones, (short)0, acc, false, false);
  }

  // Every column of D holds the row's sum of squares. Lane 0 owns rows
  // rowBase+0..7 (VGPR 0..7), lane 16 owns rows rowBase+8..15.
  if (lane == 0 || lane == 16) {
    const int base = rowBase + ((lane >> 4) << 3);
#pragma unroll
    for (int r = 0; r < 8; ++r) {
      float n = sqrtf(acc[r]);
      norms[base + r] = fmaxf(n, 1e-8f);  // matches jnp.clip(norm, 1e-8)
    }
  }
}

// ---------------------------------------------------------------------------
// Branchless nearest-centroid on a sorted table in LDS.
// 8-step lower_bound (result in [0, 255] = first index with cent[i] >= v,
// clamped), then neighbor compare; tie goes to the lower index to match
// jnp.argmin's first-occurrence rule.
// ---------------------------------------------------------------------------
__device__ __forceinline__ int pq_nearest(const float* __restrict__ cent,
                                          float v) {
  int pos = 0;
#pragma unroll
  for (int b = 128; b >= 1; b >>= 1) {
    int t = pos + b;
    pos = (cent[t - 1] < v) ? t : pos;
  }
  // pos in [0, 255]; pos == 256 is impossible since the loop sums to 255 max.
  int lo = pos - 1;
  if (pos > 0) {
    float dlo = v - cent[lo];     // >= 0
    float dhi = cent[pos] - v;    // cent[pos] >= v when pos < 256 hit
    pos = (dlo <= dhi) ? lo : pos;
  }
  return pos;
}

// ---------------------------------------------------------------------------
// Kernel 2: rotate -> quantize -> inverse rotate -> rescale.
//   One block = 256 threads = half a row (256 groups); each thread owns one
//   2-vector rotation group -> fully coalesced float2 traffic.
// ---------------------------------------------------------------------------
__global__ __launch_bounds__(256) void PQ_quant(
    const float* __restrict__ x,
    const float* __restrict__ cent_g,
    const float* __restrict__ rot2,
    const float* __restrict__ norms,
    float* __restrict__ xhat,
    float* __restrict__ idx_out) {
  __shared__ float cent[PQ_K];

  const int tid = threadIdx.x;
  cent[tid] = cent_g[tid];  // 256 threads stage the whole codebook
  __syncthreads();

  const int row = blockIdx.x >> 1;                 // 2 blocks per row
  const int g   = ((blockIdx.x & 1) << 8) + tid;   // group id 0..511
  const size_t off = (size_t)row * PQ_D + 2 * (size_t)g;

  const float nrm = norms[row];      // already clipped >= 1e-8
  const float inv = 1.0f / nrm;

  const float2 v  = *(const float2*)(x + off);
  const float2 cs = *(const float2*)(rot2 + 2 * (size_t)g);
  const float cth = cs.x;
  const float sth = cs.y;

  // normalize + forward rotation
  const float v0 = v.x * inv;
  const float v1 = v.y * inv;
  const float r0 = cth * v0 - sth * v1;
  const float r1 = sth * v0 + cth * v1;

  // per-scalar nearest centroid (sorted codebook, binary search in LDS)
  const int i0 = pq_nearest(cent, r0);
  const int i1 = pq_nearest(cent, r1);
  const float q0 = cent[i0];
  const float q1 = cent[i1];

  // inverse rotation + rescale
  const float o0 = ( cth * q0 + sth * q1) * nrm;
  const float o1 = (-sth * q0 + cth * q1) * nrm;

  float2 outv; outv.x = o0; outv.y = o1;
  *(float2*)(xhat + off) = outv;

  float2 outi; outi.x = (float)i0; outi.y = (float)i1;
  *(float2*)(idx_out + off) = outi;
}

// ---------------------------------------------------------------------------
// Host-side launcher. Inputs (setup_inputs order): x[2048*1024] f32,
// centroids[256] f32, rot2[512*2] f32. Output: x_hat (2048*1024) then
// indices (2048*1024), concatenated flat. Workspace: 2048 floats of norms.
// ---------------------------------------------------------------------------
extern "C" void kernel_launch(void* const* d_in, const int* in_sizes, int n_in,
                              void* d_out, int out_size, void* d_ws,
                              size_t ws_size, hipStream_t stream) {
  (void)in_sizes; (void)n_in; (void)out_size; (void)ws_size;

  const float* x     = (const float*)d_in[0];
  const float* cent  = (const float*)d_in[1];
  const float* rot2  = (const float*)d_in[2];

  float* xhat  = (float*)d_out;
  float* idxf  = (float*)d_out + (size_t)PQ_N * PQ_D;
  float* norms = (float*)d_ws;  // 2048 floats

  // Phase 1: row norms via FP32 WMMA. 128 single-wave workgroups x 16 rows.
  PQ_norms_wmma<<<PQ_N / 16, 32, 0, stream>>>(x, norms);

  // Phase 2: quantization. 2 blocks per row, 256 threads each.
  PQ_quant<<<PQ_N * 2, 256, 0, stream>>>(x, cent, rot2, norms, xhat, idxf);
}